// GroupConv2d_64149631533714
// MI455X (gfx1250) — compile-verified
//
#include <hip/hip_runtime.h>

typedef float v2f __attribute__((ext_vector_type(2)));
typedef float v8f __attribute__((ext_vector_type(8)));

#define N_IMG   16
#define IN_CH   1024
#define OUT_CH  1024
#define NGROUP  512
#define H_IN    56
#define W_IN    56
#define H_OUT   54
#define W_OUT   54
#define KTAPS   18                    // 2 ic * 3 * 3
#define PLANE   (H_IN * W_IN)         // 3136
#define OPLANE  (H_OUT * W_OUT)       // 2916
#define ROWS_PER_WAVE 6
#define NCHUNK  (H_OUT / ROWS_PER_WAVE)   // 9

// Compile-time element offset of tap k inside the group's input block.
__host__ __device__ constexpr int offk(int k) {
    return (k / 9) * PLANE + ((k % 9) / 3) * W_IN + ((k % 9) % 3);
}

// One wave32 (one 32-thread workgroup -> all strip coords scalar) computes a
// 2(oc) x 16(px) x (4 xt) x ROWS_PER_WAVE strip of one group's output.
// Per 16-px tile: 5 chained V_WMMA_F32_16X16X4_F32 (K=18 padded to 20; the
// pad columns are zeroed in the A registers once, so B pad loads are dummies).
//   A (16x4): M = oc (rows 0..1 real), lanes 0-15 K=4s+{0,1}, 16-31 K=4s+{2,3}
//   B (4x16): N = pixel = lane%16, same per-half K split
//   D: VGPR0 lanes0-15 = oc0 row, VGPR1 = oc1 row -> coalesced b32 stores.
__global__ __launch_bounds__(32)
void GroupConv2d_wmma_kernel(const float* __restrict__ x,
                             const float* __restrict__ w,
                             float* __restrict__ out) {
    // ---- scalar strip decode (blockIdx only -> SGPRs) ----
    int wid = blockIdx.x;
    const int chunk = wid % NCHUNK;  wid /= NCHUNK;
    const int g     = wid % NGROUP;  wid /= NGROUP;
    const int n     = wid;
    const int oy0   = chunk * ROWS_PER_WAVE;

    const int lane = threadIdx.x;
    const int m    = lane & 15;      // oc row (A) / pixel col (B)
    const int hi   = lane >> 4;      // 0: K=4s+{0,1}, 1: K=4s+{2,3}

    // ---- load group weights into A registers once (pad K=18,19 stays 0) ----
    const float* __restrict__ wg = w + (size_t)(2 * g) * KTAPS;   // scalar
    v2f A[5];
#pragma unroll
    for (int s = 0; s < 5; ++s) { A[s][0] = 0.f; A[s][1] = 0.f; }
    if (m < 2) {
        const float* __restrict__ wr = wg + m * KTAPS;            // 8B aligned
#pragma unroll
        for (int s = 0; s < 4; ++s) {
            const int kk = hi ? (4 * s + 2) : (4 * s);            // even
            A[s] = *(const v2f*)(wr + kk);
        }
        if (!hi) A[4] = *(const v2f*)(wr + 16);   // K=16,17 ; hi half stays 0
    }

    // ---- per-lane B element offsets, computed once; pad -> dummy tap 17 ----
    int voff[10];
#pragma unroll
    for (int s = 0; s < 5; ++s) {
        const int klo0 = 4 * s, klo1 = 4 * s + 1;
        const int khi0 = (4 * s + 2 < KTAPS) ? 4 * s + 2 : KTAPS - 1;
        const int khi1 = (4 * s + 3 < KTAPS) ? 4 * s + 3 : KTAPS - 1;
        voff[2 * s]     = m + (hi ? offk(khi0) : offk(klo0));
        voff[2 * s + 1] = m + (hi ? offk(khi1) : offk(klo1));
    }

    // scalar row base pointers, bumped per output row
    const float* __restrict__ xrow =
        x + ((size_t)n * IN_CH + 2 * g) * PLANE + (size_t)oy0 * W_IN;
    float* __restrict__ orow =
        out + (((size_t)n * OUT_CH + 2 * g) * H_OUT + oy0) * W_OUT;

    for (int r = 0; r < ROWS_PER_WAVE; ++r) {
        float r0[4], r1[4];
#pragma unroll
        for (int xt = 0; xt < 4; ++xt) {
            const int ox0 = (xt == 3) ? (W_OUT - 16) : (xt * 16); // 0,16,32,38
            v8f c;
            {   // step 0: inline-zero C
                v2f B; B[0] = xrow[voff[0] + ox0]; B[1] = xrow[voff[1] + ox0];
                v8f z = {};
                c = __builtin_amdgcn_wmma_f32_16x16x4_f32(false, A[0], false, B,
                                                          (short)0, z, false, false);
            }
#pragma unroll
            for (int s = 1; s < 5; ++s) {
                v2f B;
                B[0] = xrow[voff[2 * s] + ox0];
                B[1] = xrow[voff[2 * s + 1] + ox0];
                c = __builtin_amdgcn_wmma_f32_16x16x4_f32(false, A[s], false, B,
                                                          (short)0, c, false, false);
            }
            r0[xt] = c[0];    // oc0 row (lanes 0-15, N = lane = pixel)
            r1[xt] = c[1];    // oc1 row
        }
        // one exec-toggle per row: 8 coalesced b32 stores
        if (lane < 16) {
#pragma unroll
            for (int xt = 0; xt < 4; ++xt) {
                const int ox0 = (xt == 3) ? (W_OUT - 16) : (xt * 16);
                orow[ox0 + lane]          = r0[xt];
                orow[OPLANE + ox0 + lane] = r1[xt];
            }
        }
        xrow += W_IN;
        orow += W_OUT;
    }
}

extern "C" void kernel_launch(void* const* d_in, const int* in_sizes, int n_in,
                              void* d_out, int out_size, void* d_ws, size_t ws_size,
                              hipStream_t stream) {
    (void)in_sizes; (void)n_in; (void)out_size; (void)d_ws; (void)ws_size;
    const float* x = (const float*)d_in[0];
    const float* w = (const float*)d_in[1];
    float* out = (float*)d_out;

    // one 6-row strip per 32-thread block: 16 * 512 * 9 = 73,728 blocks
    const int blocks = N_IMG * NGROUP * NCHUNK;
    GroupConv2d_wmma_kernel<<<blocks, 32, 0, stream>>>(x, w, out);
}